// QConv2D_MF_27187142984005
// MI455X (gfx1250) — compile-verified
//
#include <hip/hip_runtime.h>
#include <hip/hip_bf16.h>

// ---------------------------------------------------------------------------
// QConv (quanvolution) for MI455X / gfx1250.
//
// Math restructuring: the quantum circuit is a fixed linear operator per
// filter, M_f in C^{64x64}.  psi0 is real, so
//     out_re = psi0 @ M_re^T ,  out_im = psi0 @ M_im^T
// i.e. two real (123008 x 64) x (64 x 64) GEMMs per filter -> fp32 WMMA
// (V_WMMA_F32_16X16X4_F32, keeps complex64-grade precision).
// Then probs = re^2+im^2 and Z-expectations via +/-1 signed sums.
// ---------------------------------------------------------------------------

#define NQ   6
#define DIM  64
#define FNUM 4
#define LNUM 2
#define OH   31          // (64-4)/2+1
#define NPB  961         // OH*OH
#define NST  123008      // 128*961

typedef __attribute__((ext_vector_type(2))) float v2f;
typedef __attribute__((ext_vector_type(8))) float v8f;

// ---------------------------------------------------------------------------
// Kernel 1: build M_f (64x64 complex) per filter by evolving the 64 basis
// columns through the circuit.  One block per filter, one thread per column;
// every gate acts only within a column -> zero barriers needed.
// ---------------------------------------------------------------------------
__global__ __launch_bounds__(64) void build_circuit_mats(
    const float* __restrict__ w,   // (F, L, NQ, 3)
    float*       __restrict__ Mg)  // (F, 2, 64, 64) row-major [j][k]
{
  __shared__ float Sre[DIM * DIM];   // S[d][col], col == threadIdx.x
  __shared__ float Sim[DIM * DIM];
  const int t = threadIdx.x;
  const int f = blockIdx.x;

  for (int d = 0; d < DIM; ++d) {
    Sre[d * DIM + t] = (d == t) ? 1.0f : 0.0f;
    Sim[d * DIM + t] = 0.0f;
  }

  for (int it = 0; it < LNUM; ++it) {
    for (int l = 0; l < LNUM; ++l) {
      // --- 6 single-qubit rotations ---
      for (int wq = 0; wq < NQ; ++wq) {
        const float* wf = &w[((f * LNUM + l) * NQ + wq) * 3];
        float phi = wf[0], th = wf[1], om = wf[2];
        float c = __cosf(th * 0.5f), s = __sinf(th * 0.5f);
        float ap = (phi + om) * 0.5f, am = (phi - om) * 0.5f;
        float cap = __cosf(ap), sap = __sinf(ap);
        float cam = __cosf(am), sam = __sinf(am);
        // U = [[e^{-i ap} c, -e^{i am} s], [e^{-i am} s, e^{i ap} c]]
        float u00r =  cap * c, u00i = -sap * c;
        float u01r = -cam * s, u01i = -sam * s;
        float u10r =  cam * s, u10i = -sam * s;
        float u11r =  cap * c, u11i =  sap * c;
        int bit  = 1 << (5 - wq);     // qubit wq <-> bit (5-wq)
        int mask = bit - 1;
        for (int p = 0; p < 32; ++p) {
          int d0 = ((p & ~mask) << 1) | (p & mask);
          int d1 = d0 | bit;
          float ar = Sre[d0 * DIM + t], ai = Sim[d0 * DIM + t];
          float br = Sre[d1 * DIM + t], bi = Sim[d1 * DIM + t];
          Sre[d0 * DIM + t] = u00r * ar - u00i * ai + u01r * br - u01i * bi;
          Sim[d0 * DIM + t] = u00r * ai + u00i * ar + u01r * bi + u01i * br;
          Sre[d1 * DIM + t] = u10r * ar - u10i * ai + u11r * br - u11i * bi;
          Sim[d1 * DIM + t] = u10r * ai + u10i * ar + u11r * bi + u11i * br;
        }
      }
      // --- ring of CNOTs, shift r = l % (NQ-1) + 1 ---
      int r = l % (NQ - 1) + 1;
      for (int wq = 0; wq < NQ; ++wq) {
        int cbit = 1 << (5 - wq);
        int tbit = 1 << (5 - ((wq + r) % NQ));
        for (int d = 0; d < DIM; ++d) {
          if ((d & cbit) && !(d & tbit)) {
            int d2 = d | tbit;
            float tr = Sre[d * DIM + t];
            Sre[d * DIM + t] = Sre[d2 * DIM + t]; Sre[d2 * DIM + t] = tr;
            float ti = Sim[d * DIM + t];
            Sim[d * DIM + t] = Sim[d2 * DIM + t]; Sim[d2 * DIM + t] = ti;
          }
        }
      }
    }
  }

  // M[j][k]: out[j] = sum_k M[j][k] psi[k]; S column t is basis k=t.
  for (int d = 0; d < DIM; ++d) {
    Mg[(f * 2 + 0) * (DIM * DIM) + d * DIM + t] = Sre[d * DIM + t];
    Mg[(f * 2 + 1) * (DIM * DIM) + d * DIM + t] = Sim[d * DIM + t];
  }
}

// ---------------------------------------------------------------------------
// Kernel 2: fused patch-extract + normalize + complex GEMM (f32 WMMA) +
// |.|^2 + signed Z reduction.  grid=(1922, F), block=128 (4 waves).
// Each wave owns 16 states and all 64 output columns:
//   accumulators: 4 j-tiles x {re,im} x v8f = 64 VGPRs,
//   16 K-steps x 4 tiles x 2 = 128 v_wmma_f32_16x16x4_f32 per wave.
// LDS: M_re/M_im padded (stride 66) = 33 KB + 4x16x66 patch = 17 KB.
// ---------------------------------------------------------------------------
#define MPAD 66

__global__ __launch_bounds__(128) void qconv_wmma(
    const float* __restrict__ x,    // (128, 4, 64, 64)
    const float* __restrict__ Mg,   // (F, 2, 64, 64)
    float*       __restrict__ out)  // (128, F*NQ, 31, 31)
{
  __shared__ float Ml[2][DIM * MPAD];       // padded row-major M_re / M_im
  __shared__ float patch[4][16 * MPAD];     // per-wave 16 states x 64 (reused for probs)

  const int tid = threadIdx.x;
  const int f   = blockIdx.y;

  // cooperative copy of this filter's M_re / M_im into padded LDS
  for (int i = tid; i < 2 * DIM * DIM; i += 128) {
    int part = i >> 12;            // 0 = re, 1 = im
    int j    = (i >> 6) & 63;
    int k    = i & 63;
    Ml[part][j * MPAD + k] = Mg[f * (2 * DIM * DIM) + i];
  }
  __syncthreads();

  const int wave  = tid >> 5;
  const int lane  = tid & 31;
  const int srow  = lane & 15;     // A-row (state) / B-row (j) within tile
  const int half  = lane >> 4;     // K-pair selector per WMMA layout
  const int sbase = (blockIdx.x * 4 + wave) * 16;

  float* pat = &patch[wave][0];

  // ---- patch extraction: 16 states x 64 features, uniform 32-iter loop ----
  for (int i = lane; i < 16 * DIM; i += 32) {
    int s = i >> 6, d = i & 63;
    int n = sbase + s;
    int b = n / NPB;  int rem = n - b * NPB;
    int oy = rem / OH; int ox = rem - oy * OH;
    int c  = d >> 4;  int ky = (d >> 2) & 3; int kx = d & 3;
    pat[s * MPAD + d] = x[((b * 4 + c) * 64 + (oy * 2 + ky)) * 64 + (ox * 2 + kx)];
  }

  // ---- L2 norm of state srow: each half-wave sums 32 features ----
  float part = 0.0f;
  for (int d = half * 32; d < half * 32 + 32; ++d) {
    float v = pat[srow * MPAD + d];
    part += v * v;
  }
  part += __shfl_xor(part, 16, 32);
  float rnorm = __frsqrt_rn(part);

  // ---- complex GEMM: out[s][j] = sum_k psi[s][k] * M[j][k] ----
  v8f accre[4] = {};
  v8f accim[4] = {};
  for (int kk = 0; kk < 16; ++kk) {
    int kb = kk * 4 + half * 2;
    // A (16x4 f32): lane holds psi[srow][kb], psi[srow][kb+1]
    v2f a = *(const v2f*)&pat[srow * MPAD + kb];
    a *= rnorm;
    for (int t = 0; t < 4; ++t) {
      int j = t * 16 + srow;
      // B (4x16 f32): lane holds M[j][kb], M[j][kb+1]  (B[k][j] = M[j][k])
      v2f bre = *(const v2f*)&Ml[0][j * MPAD + kb];
      v2f bim = *(const v2f*)&Ml[1][j * MPAD + kb];
      accre[t] = __builtin_amdgcn_wmma_f32_16x16x4_f32(
          false, a, false, bre, (short)0, accre[t], false, false);
      accim[t] = __builtin_amdgcn_wmma_f32_16x16x4_f32(
          false, a, false, bim, (short)0, accim[t], false, false);
    }
  }

  // ---- probs = re^2 + im^2, scatter to LDS (reuse patch buffer) ----
  // C/D layout: VGPR r -> state r (lanes 0-15) / r+8 (lanes 16-31), col = lane%16
  for (int t = 0; t < 4; ++t) {
    v8f pre = accre[t], pim = accim[t];
    for (int r = 0; r < 8; ++r) {
      float p = pre[r] * pre[r] + pim[r] * pim[r];
      pat[(r + half * 8) * MPAD + (t * 16 + srow)] = p;
    }
  }

  // ---- Z expectations: exp[s][q] = sum_j sign(q,j) * prob[s][j] ----
  // 96 outputs per wave, uniform 3-iter loop; write transposed layout.
  for (int i = lane; i < 16 * NQ; i += 32) {
    int s = i / NQ;
    int q = i - s * NQ;
    int shift = 5 - q;
    float sum = 0.0f;
    for (int j = 0; j < DIM; ++j) {
      float p = pat[s * MPAD + j];
      sum += ((j >> shift) & 1) ? -p : p;
    }
    int n = sbase + s;
    int b = n / NPB;  int rem = n - b * NPB;
    int oy = rem / OH; int ox = rem - oy * OH;
    out[((b * (FNUM * NQ) + f * NQ + q) * OH + oy) * OH + ox] = sum;
  }
}

// ---------------------------------------------------------------------------
extern "C" void kernel_launch(void* const* d_in, const int* in_sizes, int n_in,
                              void* d_out, int out_size, void* d_ws, size_t ws_size,
                              hipStream_t stream) {
  (void)in_sizes; (void)n_in; (void)out_size; (void)ws_size;
  const float* x = (const float*)d_in[0];   // (128,4,64,64) f32
  const float* w = (const float*)d_in[1];   // (4,2,6,3) f32
  float* out = (float*)d_out;               // (128,24,31,31) f32
  float* Mg  = (float*)d_ws;                // 4*2*64*64 floats = 128 KB scratch

  build_circuit_mats<<<dim3(FNUM), dim3(64), 0, stream>>>(w, Mg);
  // 1922 blocks * 4 waves * 16 states = 123008 = N
  qconv_wmma<<<dim3(1922, FNUM), dim3(128), 0, stream>>>(x, Mg, out);
}